// SeqAttentionBlock_57019985821780
// MI455X (gfx1250) — compile-verified
//
#include <hip/hip_runtime.h>
#include <math.h>

// Problem constants (match reference)
#define Bb_ 2
#define T_ 512
#define M_ 8
#define H_ 4
#define E_ 32
#define D_ 128
#define P_ 128
#define MH_ 32          // M_*H_
#define PAD_ 4          // LDS row pad (dwords) -> conflict-free v2f fragment reads
#define ROWW_ (D_ + PAD_)   // 132
#define SROWW_ (T_ + 4)     // 516
#define LCH_ 64         // pos rows staged per chunk
#define TB_ 4           // (b,t) rows per block in proj kernel

typedef float v2f __attribute__((ext_vector_type(2)));
typedef float v8f __attribute__((ext_vector_type(8)));

#define HAS_ASYNC_LDS __has_builtin(__builtin_amdgcn_global_load_async_to_lds_b128)

#if HAS_ASYNC_LDS
typedef int v4i_ __attribute__((vector_size(16)));
typedef __attribute__((address_space(1))) v4i_ as1_v4i;   // global
typedef __attribute__((address_space(3))) v4i_ as3_v4i;   // LDS
// One b128 per lane: copies 16 bytes from global address g to LDS address l, tracked by ASYNCcnt (no VGPR data).
__device__ __forceinline__ void async_copy16(const float* g, float* l) {
    __builtin_amdgcn_global_load_async_to_lds_b128((as1_v4i*)g, (as3_v4i*)l, 0, 0);
}
#endif

__device__ __forceinline__ void wait_async0() {
#if __has_builtin(__builtin_amdgcn_s_wait_asynccnt)
    __builtin_amdgcn_s_wait_asynccnt(0);
#else
    asm volatile("s_wait_asynccnt 0x0" ::: "memory");
#endif
}

// ---------------------------------------------------------------------------
// Kernel 1: per (b,t): q,k,v,qt projections; fold Wtd into qt -> u; Btd -> c.
//   q_ws[bt][mh*32+e], k_ws/v_ws[b][mh][t][e], u_ws[bt][mh][d], c_ws[bt][mh]
// ---------------------------------------------------------------------------
__global__ void proj_kernel(const float* __restrict__ inp,
                            const float* __restrict__ Wq, const float* __restrict__ Bq,
                            const float* __restrict__ Wk, const float* __restrict__ Bk,
                            const float* __restrict__ Wv, const float* __restrict__ Bv,
                            const float* __restrict__ Wt, const float* __restrict__ Bt,
                            const float* __restrict__ Wtd, const float* __restrict__ Btd,
                            float* __restrict__ q_ws, float* __restrict__ k_ws,
                            float* __restrict__ v_ws, float* __restrict__ u_ws,
                            float* __restrict__ c_ws)
{
    __shared__ float s_inp[TB_ * M_ * D_];
    __shared__ float s_qt [TB_ * M_ * P_];
    const int tid = threadIdx.x;
    const int bt0 = blockIdx.x * TB_;        // 4 consecutive (b,t); never spans b
    const int b   = bt0 / T_;

    {   // stage inp for 4 (b,t) rows: contiguous 4096 floats
        const float4* src = (const float4*)inp + (size_t)bt0 * (M_ * D_ / 4);
        float4* dst = (float4*)s_inp;
        for (int i = tid; i < TB_ * M_ * D_ / 4; i += 256) dst[i] = src[i];
    }
    __syncthreads();

    const float* Ws[4]  = {Wq, Wk, Wv, Wt};
    const float* Bss[4] = {Bq, Bk, Bv, Bt};

    // 4 mats * 1024 outputs, 4 bt each; mat uniform per iteration
    for (int i = 0; i < 16; ++i) {
        const int o   = tid + (i << 8);
        const int mat = o >> 10;
        const int mp  = o & 1023;           // m*128 + h*32 + e
        const int m   = mp >> 7;
        const float* wrow = Ws[mat] + (size_t)(m * P_ + (mp & 127)) * D_;
        const float* x0 = s_inp + 0 * 1024 + m * D_;
        const float* x1 = s_inp + 1 * 1024 + m * D_;
        const float* x2 = s_inp + 2 * 1024 + m * D_;
        const float* x3 = s_inp + 3 * 1024 + m * D_;
        float a0 = 0.f, a1 = 0.f, a2 = 0.f, a3 = 0.f;
        #pragma unroll 4
        for (int d = 0; d < D_; ++d) {
            const float w = wrow[d];
            a0 += w * x0[d]; a1 += w * x1[d]; a2 += w * x2[d]; a3 += w * x3[d];
        }
        const float bias = Bss[mat][mp];
        a0 += bias; a1 += bias; a2 += bias; a3 += bias;
        const float acc[TB_] = {a0, a1, a2, a3};
        const int mh = mp >> 5, e = mp & 31;
        for (int j = 0; j < TB_; ++j) {
            const int bt = bt0 + j, tt = bt & (T_ - 1);
            if (mat == 0)      q_ws[(size_t)bt * 1024 + mp] = acc[j];
            else if (mat == 1) k_ws[(((size_t)b * MH_ + mh) * T_ + tt) * E_ + e] = acc[j];
            else if (mat == 2) v_ws[(((size_t)b * MH_ + mh) * T_ + tt) * E_ + e] = acc[j];
            else               s_qt[j * 1024 + mp] = acc[j];
        }
    }
    __syncthreads();

    // u[bt][mh][d] = sum_e Wtd[d][h*32+e] * qt[bt][m*128+h*32+e]
    for (int i = 0; i < 16; ++i) {
        const int o  = tid + (i << 8);      // mh*128 + d
        const int mh = o >> 7, d = o & 127;
        const int m = mh >> 2, h = mh & 3;
        const float* wrow = Wtd + (size_t)d * P_ + h * E_;
        float a0 = 0.f, a1 = 0.f, a2 = 0.f, a3 = 0.f;
        const int qb = m * 128 + h * E_;
        #pragma unroll 4
        for (int e = 0; e < E_; ++e) {
            const float w = wrow[e];
            a0 += w * s_qt[0 * 1024 + qb + e];
            a1 += w * s_qt[1 * 1024 + qb + e];
            a2 += w * s_qt[2 * 1024 + qb + e];
            a3 += w * s_qt[3 * 1024 + qb + e];
        }
        u_ws[(size_t)(bt0 + 0) * 4096 + o] = a0;
        u_ws[(size_t)(bt0 + 1) * 4096 + o] = a1;
        u_ws[(size_t)(bt0 + 2) * 4096 + o] = a2;
        u_ws[(size_t)(bt0 + 3) * 4096 + o] = a3;
    }
    if (tid < MH_) {
        const int m = tid >> 2, h = tid & 3;
        for (int j = 0; j < TB_; ++j) {
            float acc = 0.f;
            for (int e = 0; e < E_; ++e)
                acc += Btd[h * E_ + e] * s_qt[j * 1024 + m * 128 + h * E_ + e];
            c_ws[(size_t)(bt0 + j) * MH_ + tid] = acc;
        }
    }
}

// ---------------------------------------------------------------------------
// Kernel 2: one workgroup per (b,t). Bias GEMM via v_wmma_f32_16x16x4_f32 with
// double-buffered async global->LDS staging of pos; fused QK^T + mask +
// softmax + weight*V.
// ---------------------------------------------------------------------------
__global__ void attn_kernel(const float* __restrict__ pos,
                            const unsigned char* __restrict__ mask,
                            const float* __restrict__ q_ws,
                            const float* __restrict__ k_ws,
                            const float* __restrict__ v_ws,
                            const float* __restrict__ u_ws,
                            const float* __restrict__ c_ws,
                            float* __restrict__ out)
{
    extern __shared__ float smem[];
    float* s_u    = smem;                        // 32 x 132
    float* s_pos0 = s_u    + MH_ * ROWW_;        // 64 x 132 (buffer 0)
    float* s_pos1 = s_pos0 + LCH_ * ROWW_;       // 64 x 132 (buffer 1)
    float* s_S    = s_pos1 + LCH_ * ROWW_;       // 32 x 516 (bias -> scores -> weights)
    float* s_q    = s_S    + MH_ * SROWW_;       // 1024
    float* s_c    = s_q    + 1024;               // 32

    const int bt = blockIdx.x;
    const int b = bt / T_, t = bt % T_;
    const int tid = threadIdx.x, lane = tid & 31, wave = tid >> 5;
    const float scale = 0.1767766952966369f;     // 1/sqrt(32)

    // stage u (padded rows), q, c
    for (int idx = tid; idx < MH_ * (D_ / 4); idx += 256) {
        const int r = idx >> 5, s = idx & 31;
        float4 v4 = ((const float4*)(u_ws + (size_t)bt * 4096 + r * D_))[s];
        *(float4*)&s_u[r * ROWW_ + s * 4] = v4;
    }
    ((float4*)s_q)[tid] = ((const float4*)(q_ws + (size_t)bt * 1024))[tid];
    if (tid < MH_) s_c[tid] = c_ws[(size_t)bt * MH_ + tid];
    __syncthreads();

    // ---- bias: bias[l,mh] = sum_d pos[b,t,l,d] * u[d,mh], only l <= t ----
    const float* posbt = pos + (size_t)bt * T_ * D_;
    const int nch = (t >> 6) + 1;
    const int ltile = wave >> 1, mht = wave & 1;       // 8 waves -> 4x2 tiles
    const int frow = lane & 15, koff = (lane >> 4) << 1;
    const int mh_c = (mht << 4) + frow;                // C-tile column (mh)

#if HAS_ASYNC_LDS
    {   // prologue: async-stage chunk 0 into buffer 0
        for (int idx = tid; idx < LCH_ * 32; idx += 256) {
            const int r = idx >> 5, s = idx & 31;
            async_copy16(posbt + (size_t)r * D_ + s * 4, s_pos0 + r * ROWW_ + s * 4);
        }
    }
#endif

    for (int c = 0; c < nch; ++c) {
        const int l0 = c << 6;
        float* buf = (c & 1) ? s_pos1 : s_pos0;
#if HAS_ASYNC_LDS
        wait_async0();          // our share of chunk c landed in LDS
        __syncthreads();        // everyone's share landed
        if (c + 1 < nch) {      // kick off chunk c+1 into the other buffer
            float* nbuf = (c & 1) ? s_pos0 : s_pos1;
            for (int idx = tid; idx < LCH_ * 32; idx += 256) {
                const int r = idx >> 5, s = idx & 31;
                async_copy16(posbt + (size_t)(l0 + LCH_ + r) * D_ + s * 4,
                             nbuf + r * ROWW_ + s * 4);
            }
        }
#else
        for (int idx = tid; idx < LCH_ * 32; idx += 256) {
            const int r = idx >> 5, s = idx & 31;
            float4 v4 = ((const float4*)(posbt + (size_t)(l0 + r) * D_))[s];
            *(float4*)&buf[r * ROWW_ + s * 4] = v4;
        }
        if (c + 1 < nch) {      // prefetch next chunk (global_prefetch_b8)
            for (int idx = tid; idx < LCH_ * 4; idx += 256) {
                const int r = idx >> 2, s = idx & 3;
                __builtin_prefetch(posbt + (size_t)(l0 + LCH_ + r) * D_ + s * 32, 0, 3);
            }
        }
        __syncthreads();
#endif

        v8f acc = {0.f, 0.f, 0.f, 0.f, 0.f, 0.f, 0.f, 0.f};
        #pragma unroll
        for (int kk = 0; kk < D_; kk += 4) {
            v2f a  = *(const v2f*)&buf [((ltile << 4) + frow) * ROWW_ + kk + koff];
            v2f bb = *(const v2f*)&s_u [((mht   << 4) + frow) * ROWW_ + kk + koff];
            acc = __builtin_amdgcn_wmma_f32_16x16x4_f32(
                false, a, false, bb, (short)0, acc, false, false);
        }
        // scatter C tile: vgpr r -> l = l0 + ltile*16 + (lane/16)*8 + r, col mh_c
        const int lbase = l0 + (ltile << 4) + ((lane >> 4) << 3);
        #pragma unroll
        for (int r = 0; r < 8; ++r) s_S[mh_c * SROWW_ + lbase + r] = acc[r];
        __syncthreads();        // bias written; pos buffer reads finished
    }

    // ---- scores: S = scale*(q.k + bias + c), causal + key-padding mask ----
    for (int mi = 0; mi < 4; ++mi) {
        const int mh = wave + (mi << 3), m = mh >> 2;
        const float* kb = k_ws + ((size_t)b * MH_ + mh) * T_ * E_;
        const float* qv = s_q + mh * E_;
        const float cc = s_c[mh];
        for (int l = lane; l <= t; l += 32) {
            const float* kr = kb + (size_t)l * E_;
            float qk = 0.f;
            #pragma unroll
            for (int e = 0; e < E_; e += 4)
                qk += qv[e]*kr[e] + qv[e+1]*kr[e+1] + qv[e+2]*kr[e+2] + qv[e+3]*kr[e+3];
            float sc = scale * (qk + s_S[mh * SROWW_ + l] + cc);
            if (!mask[((size_t)b * T_ + l) * M_ + m]) sc = -INFINITY;
            s_S[mh * SROWW_ + l] = sc;
        }
    }
    __syncthreads();

    // ---- softmax per mh row over l in [0, t] ----
    for (int mi = 0; mi < 4; ++mi) {
        const int mh = wave + (mi << 3);
        float mx = -INFINITY;
        for (int l = lane; l <= t; l += 32) mx = fmaxf(mx, s_S[mh * SROWW_ + l]);
        for (int off = 16; off; off >>= 1) mx = fmaxf(mx, __shfl_xor(mx, off));
        float sum = 0.f;
        for (int l = lane; l <= t; l += 32) {
            const float w = __expf(s_S[mh * SROWW_ + l] - mx);
            s_S[mh * SROWW_ + l] = w;
            sum += w;
        }
        for (int off = 16; off; off >>= 1) sum += __shfl_xor(sum, off);
        const float inv = 1.0f / sum;
        for (int l = lane; l <= t; l += 32) s_S[mh * SROWW_ + l] *= inv;
    }
    __syncthreads();

    // ---- out[mh,e] = sum_{l<=t} w[mh,l] * v[b,mh,l,e] ----
    float* outbt = out + (size_t)bt * (M_ * P_);
    for (int o = tid; o < M_ * P_; o += 256) {
        const int mh = o >> 5, e = o & 31;
        const float* vb = v_ws + ((size_t)b * MH_ + mh) * T_ * E_ + e;
        float acc = 0.f;
        for (int l = 0; l <= t; ++l) acc += s_S[mh * SROWW_ + l] * vb[(size_t)l * E_];
        outbt[o] = acc;
    }
}

// ---------------------------------------------------------------------------
extern "C" void kernel_launch(void* const* d_in, const int* in_sizes, int n_in,
                              void* d_out, int out_size, void* d_ws, size_t ws_size,
                              hipStream_t stream) {
    (void)in_sizes; (void)n_in; (void)out_size; (void)ws_size;
    const float* inp = (const float*)d_in[0];
    const float* pos = (const float*)d_in[1];
    const unsigned char* mask = (const unsigned char*)d_in[2];
    const float* Wq = (const float*)d_in[3];  const float* Bq = (const float*)d_in[4];
    const float* Wk = (const float*)d_in[5];  const float* Bk = (const float*)d_in[6];
    const float* Wv = (const float*)d_in[7];  const float* Bv = (const float*)d_in[8];
    const float* Wt = (const float*)d_in[9];  const float* Bt = (const float*)d_in[10];
    const float* Wtd = (const float*)d_in[11]; const float* Btd = (const float*)d_in[12];

    float* q_ws = (float*)d_ws;                 // 1,048,576 f
    float* k_ws = q_ws + 1048576;               // 1,048,576 f
    float* v_ws = k_ws + 1048576;               // 1,048,576 f
    float* u_ws = v_ws + 1048576;               // 4,194,304 f
    float* c_ws = u_ws + 4194304;               //    32,768 f  (~29.5 MB total)

    proj_kernel<<<(Bb_ * T_) / TB_, 256, 0, stream>>>(
        inp, Wq, Bq, Wk, Bk, Wv, Bv, Wt, Bt, Wtd, Btd,
        q_ws, k_ws, v_ws, u_ws, c_ws);

    const size_t smem = (size_t)(MH_ * ROWW_ + 2 * LCH_ * ROWW_ + MH_ * SROWW_
                                 + 1024 + MH_) * sizeof(float);   // ~151 KB
    (void)hipFuncSetAttribute((const void*)attn_kernel,
                              hipFuncAttributeMaxDynamicSharedMemorySize, (int)smem);
    attn_kernel<<<Bb_ * T_, 256, smem, stream>>>(
        pos, mask, q_ws, k_ws, v_ws, u_ws, c_ws, (float*)d_out);
}